// Gcs_loss_47278999994715
// MI455X (gfx1250) — compile-verified
//
#include <hip/hip_runtime.h>

typedef __attribute__((ext_vector_type(2))) float v2f;
typedef __attribute__((ext_vector_type(8))) float v8f;

#define BDIM  32
#define PDIM  1024
#define TWO_P 2048

// One workgroup (1024 threads = 32 wave32 waves) computes the whole loss:
//   per half: v(b,i) = pred - truth  (B=32 x P=1024)
//   dist^2[i] = S2 + P*n2[i] - 2*cross[i],  cross = V^T s  via WMMA f32 16x16x4
__global__ __launch_bounds__(1024) void gcs_loss_kernel(const float* __restrict__ pred,
                                                        const float* __restrict__ truth,
                                                        float* __restrict__ out) {
    __shared__ float s_row[BDIM];   // s[b] = sum_i v(b,i)
    __shared__ float s_S2;          // total sum of squares (per half)
    __shared__ float s_total;       // accumulated sum of dists (both halves)

    const int  tid  = threadIdx.x;      // 0..1023 == column index within the half
    const int  lane = tid & 31;
    const bool hi   = (lane >= 16);

    if (tid == 0) s_total = 0.0f;

    #pragma unroll 1
    for (int half = 0; half < 2; ++half) {
        __syncthreads();
        if (tid < BDIM) s_row[tid] = 0.0f;
        if (tid == BDIM) s_S2 = 0.0f;
        __syncthreads();

        const float* pbase = pred  + half * PDIM;
        const float* tbase = truth + half * PDIM;

        // Each thread owns one column of v: cv[b] = v(b, tid)
        float cv[BDIM];
        float n2 = 0.0f;
        #pragma unroll
        for (int b = 0; b < BDIM; ++b) {
            float v = pbase[b * TWO_P + tid] - tbase[b * TWO_P + tid];
            cv[b] = v;
            n2 = fmaf(v, v, n2);
        }

        // Row sums s[b] across all 1024 columns: wave butterfly + LDS atomic
        #pragma unroll
        for (int b = 0; b < BDIM; ++b) {
            float v = cv[b];
            #pragma unroll
            for (int off = 16; off > 0; off >>= 1)
                v += __shfl_xor(v, off, 32);
            if (lane == 0) atomicAdd(&s_row[b], v);
        }
        atomicAdd(&s_S2, n2);
        __syncthreads();

        const float S2 = s_S2;

        // cross[i] = sum_b v(b,i)*s[b] for this wave's 32 columns via WMMA.
        // A (16x4 f32 layout): lanes 0-15 hold K=0(VGPR0),1(VGPR1); lanes 16-31 hold K=2,3.
        // B (4x16) mirrors A with N in place of M; here B[k][n] = s[b] replicated over n.
        v8f c0 = {};  // columns base+0..15   (A sourced from lanes 0..15)
        v8f c1 = {};  // columns base+16..31  (A sourced from lanes 16..31)
        const int srcA = (lane & 15);
        const int srcB = 16 + (lane & 15);
        #pragma unroll
        for (int kc = 0; kc < 8; ++kc) {   // k-chunks of 4 over the 32 batch rows
            const float y0 = cv[4 * kc + 0];
            const float y1 = cv[4 * kc + 1];
            const float y2 = cv[4 * kc + 2];
            const float y3 = cv[4 * kc + 3];

            v2f bb;
            bb.x = s_row[4 * kc + (hi ? 2 : 0)];
            bb.y = s_row[4 * kc + (hi ? 3 : 1)];

            // tile 0: rows M=0..15 are columns owned by lanes 0..15
            float t0 = __shfl(y0, srcA, 32);
            float t1 = __shfl(y1, srcA, 32);
            float t2 = __shfl(y2, srcA, 32);
            float t3 = __shfl(y3, srcA, 32);
            v2f a0; a0.x = hi ? t2 : t0; a0.y = hi ? t3 : t1;
            c0 = __builtin_amdgcn_wmma_f32_16x16x4_f32(false, a0, false, bb,
                                                       (short)0, c0, false, false);

            // tile 1: rows are columns owned by lanes 16..31
            float u0 = __shfl(y0, srcB, 32);
            float u1 = __shfl(y1, srcB, 32);
            float u2 = __shfl(y2, srcB, 32);
            float u3 = __shfl(y3, srcB, 32);
            v2f a1; a1.x = hi ? u2 : u0; a1.y = hi ? u3 : u1;
            c1 = __builtin_amdgcn_wmma_f32_16x16x4_f32(false, a1, false, bb,
                                                       (short)0, c1, false, false);
        }

        // D layout: VGPR p -> (M=p, lanes 0-15) / (M=p+8, lanes 16-31).
        // Column (base + M)'s n2 lives in lane M (tile0) / lane 16+M (tile1).
        float accw = 0.0f;
        const int moff = hi ? 8 : 0;
        #pragma unroll
        for (int p = 0; p < 8; ++p) {
            float n2a = __shfl(n2, p + moff, 32);
            float n2b = __shfl(n2, 16 + p + moff, 32);
            float da  = S2 + 1024.0f * n2a - 2.0f * c0[p];
            float db  = S2 + 1024.0f * n2b - 2.0f * c1[p];
            accw += sqrtf(fmaxf(da, 0.0f)) + sqrtf(fmaxf(db, 0.0f));
        }
        if (lane == 0 || lane == 16) atomicAdd(&s_total, accw);
    }

    __syncthreads();
    if (tid == 0) out[0] = s_total * (1.0f / 64.0f);   // / (B * 2)
}

extern "C" void kernel_launch(void* const* d_in, const int* in_sizes, int n_in,
                              void* d_out, int out_size, void* d_ws, size_t ws_size,
                              hipStream_t stream) {
    (void)in_sizes; (void)n_in; (void)out_size; (void)d_ws; (void)ws_size;
    const float* pred  = (const float*)d_in[0];
    const float* truth = (const float*)d_in[1];
    float* out = (float*)d_out;
    gcs_loss_kernel<<<dim3(1), dim3(1024), 0, stream>>>(pred, truth, out);
}